// DuplicateRemovalLayer_10806137717519
// MI455X (gfx1250) — compile-verified
//
#include <hip/hip_runtime.h>

typedef float v4f __attribute__((ext_vector_type(4)));

#define PI_F       3.14159265358979323846f
#define TWO_PI_F   6.28318530717958647692f
#define INV_2PI_F  0.15915494309189533577f
#define SENT_F     1000000.0f
#define THR_F      0.05f

// Python-style mod(x, 2*pi), then -pi.  floor-based; exact in the only region
// that can affect the (dr < 0.05) decision (see analysis).
__device__ __forceinline__ float dphi_wrap(float x) {
    float t = floorf(__fmul_rn(x, INV_2PI_F));
    float r = __builtin_fmaf(-TWO_PI_F, t, x);   // x - 2pi*t, single rounding
    return __fadd_rn(r, -PI_F);
}

// For objects [S, E): dup[j] |= exists i>j with dR(i,j) < THR, using
// sentinel-masked eta/phi exactly like the reference.
template <int S, int E>
__device__ __forceinline__ void dup_group(const float* __restrict__ v,
                                          bool* __restrict__ dup) {
#pragma unroll
    for (int j = S; j < E; ++j) {
        const bool  aj = v[3 * j] > 0.0f;
        const float ej = aj ? v[3 * j + 1] : SENT_F;
        const float pj = aj ? v[3 * j + 2] : SENT_F;
        bool d = false;
#pragma unroll
        for (int i = j + 1; i < E; ++i) {
            const bool  ai = v[3 * i] > 0.0f;
            const float ei = ai ? v[3 * i + 1] : SENT_F;
            const float pi_ = ai ? v[3 * i + 2] : SENT_F;
            const float de = __fadd_rn(ei, -ej);
            const float dp = dphi_wrap(__fadd_rn(__fadd_rn(pi_, -pj), PI_F));
            const float dr2 = __fadd_rn(__fmul_rn(de, de), __fmul_rn(dp, dp));
            d = d || (sqrtf(dr2) < THR_F);
        }
        dup[j] = dup[j] || d;
    }
}

__global__ __launch_bounds__(256)
void dup_removal_kernel(const float* __restrict__ in,
                        float* __restrict__ out,
                        int nEvents) {
    // 8 waves/block, 32 events/wave, 12 float4 per event -> 48 KB LDS.
    __shared__ v4f smem[8 * 32 * 12];

    const int lane = threadIdx.x & 31;
    const int wIn  = threadIdx.x >> 5;
    const long long waveG  = (long long)blockIdx.x * 8 + wIn;
    const long long evBase = waveG * 32;
    const long long ev     = evBase + lane;

    const bool fullWave  = (evBase + 32) <= (long long)nEvents; // wave-uniform
    const bool haveEvent = ev < (long long)nEvents;

    alignas(16) float v[48];
    const v4f* __restrict__ gin = (const v4f*)in;

    if (fullWave) {
        // --- CDNA5 async data mover: stage this wave's 32 events (6 KB) into
        // LDS with 12 coalesced b128 async loads per lane. Instruction offset
        // applies to both the LDS and global address.
        unsigned ldsOff = (unsigned)(size_t)(&smem[wIn * 384]) + (unsigned)lane * 16u;
        unsigned gOff   = (unsigned)evBase * 192u + (unsigned)lane * 16u;

#define ALOAD(OFF)                                                       \
        asm volatile("global_load_async_to_lds_b128 %0, %1, %2 offset:" #OFF \
                     :: "v"(ldsOff), "v"(gOff), "s"(in) : "memory")
        ALOAD(0);    ALOAD(512);  ALOAD(1024); ALOAD(1536);
        ALOAD(2048); ALOAD(2560); ALOAD(3072); ALOAD(3584);
        ALOAD(4096); ALOAD(4608); ALOAD(5120); ALOAD(5632);
#undef ALOAD

        asm volatile("s_wait_asynccnt 0x0" ::: "memory");

        const v4f* __restrict__ my = &smem[wIn * 384 + lane * 12];
#pragma unroll
        for (int k = 0; k < 12; ++k) ((v4f*)v)[k] = my[k];
    } else {
        if (!haveEvent) return;
#pragma unroll
        for (int k = 0; k < 12; ++k)
            ((v4f*)v)[k] = __builtin_nontemporal_load(&gin[ev * 12 + k]);
    }

    // --- duplicate flags per group: jets [0,6), e [6,9), mu [9,12), ph [12,15)
    bool dup[16];
#pragma unroll
    for (int o = 0; o < 16; ++o) dup[o] = false;
    dup_group<0, 6>(v, dup);
    dup_group<6, 9>(v, dup);
    dup_group<9, 12>(v, dup);
    dup_group<12, 15>(v, dup);
    // dup[15] stays false: MET row passes through.

    alignas(16) float ov[48];
#pragma unroll
    for (int o = 0; o < 16; ++o) {
        const bool z = dup[o];
        ov[3 * o + 0] = z ? 0.0f : v[3 * o + 0];
        ov[3 * o + 1] = z ? 0.0f : v[3 * o + 1];
        ov[3 * o + 2] = z ? 0.0f : v[3 * o + 2];
    }

    v4f* __restrict__ gout = (v4f*)out;
#pragma unroll
    for (int k = 0; k < 12; ++k)
        __builtin_nontemporal_store(((const v4f*)ov)[k], &gout[ev * 12 + k]);
}

extern "C" void kernel_launch(void* const* d_in, const int* in_sizes, int n_in,
                              void* d_out, int out_size, void* d_ws, size_t ws_size,
                              hipStream_t stream) {
    (void)n_in; (void)out_size; (void)d_ws; (void)ws_size;
    const float* x   = (const float*)d_in[0];
    float*       out = (float*)d_out;

    const int nEvents = in_sizes[0] / 48;            // 16 objects * 3 floats
    const int waves   = (nEvents + 31) / 32;         // 31250 for 1M events
    const int blocks  = (waves + 7) / 8;             // 8 waves (256 thr) / block

    dup_removal_kernel<<<blocks, 256, 0, stream>>>(x, out, nEvents);
}